// Sampler_7241314861619
// MI455X (gfx1250) — compile-verified
//
#include <hip/hip_runtime.h>
#include <math.h>
#include <limits.h>

// MI455X sampler.
// Memory-bound: floor ~11us (131MB read + 131MB write at 23.3 TB/s). Plan:
//   Pass A: async-LDS-staged stream -> online softmax max/sumexp (ASYNCcnt path)
//   Pass B: re-read (mostly L2-resident: 131MB < 192MB L2, helped by NT stores
//           for the output zero-fill) -> zero-fill probs + collect tail window
//           candidates [rowMax-Delta, rowMax] (~1k elements, k<=63 guaranteed
//           inside with retry/widen loop).
//   Then: small bitonic sort + exact top-k/top-p/min-p/sample on <=64 values.
// CDNA5-specific: global_load_async_to_lds_b128, s_wait_asynccnt, s_wait_dscnt,
// non-temporal b128 stores.

#define VOCAB 128000
#define NB    256          // threads per block (8 wave32)
#define CAP   4096         // candidate buffer capacity
#define TILE  1024         // floats per staged tile = NB * 4

typedef float floatx4 __attribute__((ext_vector_type(4)));  // native vec for NT store

__global__ __launch_bounds__(NB) void sampler_kernel(
    const float* __restrict__ logits,
    const float* __restrict__ temps,
    const int*   __restrict__ topks,
    const float* __restrict__ topps,
    const float* __restrict__ minps,
    const float* __restrict__ noise,
    int*   __restrict__ tokensOut,
    float* __restrict__ probsOut)
{
  const int r   = blockIdx.x;
  const int tid = threadIdx.x;
  const float T    = temps[r];
  const float invT = 1.0f / T;

  __shared__ __align__(16) float tile[2][TILE];   // async staging (8KB)
  __shared__ float sRedM[NB / 32];
  __shared__ float sRedS[NB / 32];
  __shared__ float sRowMax, sRowZ;
  __shared__ int   sCount;
  __shared__ float candV[CAP];                    // 16KB
  __shared__ int   candI[CAP];                    // 16KB
  __shared__ float sQ[64];
  __shared__ float sF[64];
  __shared__ int   sIdx2[64];
  __shared__ int   sNK;
  __shared__ float sTarget;
  __shared__ int   sTok;

  if (tid == 0) sTok = INT_MAX;

  const float* rowPtr = logits + (size_t)r * VOCAB;

  // ---------------- Pass A: async-staged stream -------------------------
  // online softmax stats (max & sumexp of logit*invT); no LDS atomics here.
  float m = -INFINITY, s = 0.0f;
  const int nIter = VOCAB / TILE;   // 125, exact
  {
    const unsigned vbase = (unsigned)tid * 16u;   // 16B per lane per tile
    {
      unsigned lds0 = (unsigned)(size_t)&tile[0][tid * 4];
      asm volatile("global_load_async_to_lds_b128 %0, %1, %2"
                   :: "v"(lds0), "v"(vbase), "s"(rowPtr) : "memory");
    }
    for (int it = 0; it < nIter; ++it) {
      const int buf = it & 1;
      if (it + 1 < nIter) {
        // our own ds_loads from buf^1 (previous iteration) must drain before
        // the async engine overwrites that buffer
        asm volatile("s_wait_dscnt 0x0" ::: "memory");
        unsigned ldsn = (unsigned)(size_t)&tile[buf ^ 1][tid * 4];
        unsigned voff = (unsigned)(it + 1) * (TILE * 4u) + vbase;
        asm volatile("global_load_async_to_lds_b128 %0, %1, %2"
                     :: "v"(ldsn), "v"(voff), "s"(rowPtr) : "memory");
        asm volatile("s_wait_asynccnt 0x1" ::: "memory");  // oldest done
      } else {
        asm volatile("s_wait_asynccnt 0x0" ::: "memory");
      }
      float4 x4 = *(const float4*)&tile[buf][tid * 4];
      float xs[4] = {x4.x, x4.y, x4.z, x4.w};
#pragma unroll
      for (int c = 0; c < 4; ++c) {
        float l = xs[c] * invT;
        if (l > m) { s = s * __expf(m - l) + 1.0f; m = l; }
        else       { s += __expf(l - m); }
      }
    }
  }

  // block reduction of (max, sumexp)
#pragma unroll
  for (int off = 16; off > 0; off >>= 1) {
    float m2 = __shfl_xor(m, off, 32);
    float s2 = __shfl_xor(s, off, 32);
    float M = fmaxf(m, m2);
    s = s * __expf(m - M) + s2 * __expf(m2 - M);
    m = M;
  }
  if ((tid & 31) == 0) { sRedM[tid >> 5] = m; sRedS[tid >> 5] = s; }
  __syncthreads();
  if (tid == 0) {
    float M = sRedM[0], S = sRedS[0];
    for (int w = 1; w < NB / 32; ++w) {
      float m2 = sRedM[w], s2 = sRedS[w];
      float Mn = fmaxf(M, m2);
      S = S * __expf(M - Mn) + s2 * __expf(m2 - Mn);
      M = Mn;
    }
    sRowMax = M; sRowZ = S;
  }
  __syncthreads();

  int kk = topks[r]; if (kk < 1) kk = 1; if (kk > 63) kk = 63;
  const float rawMax = sRowMax * T;   // max in raw-logit units

  // ---------------- Pass B: zero-fill output + windowed collect ---------
  float* outRow = probsOut + (size_t)r * VOCAB;
  const float4* row4 = (const float4*)rowPtr;
  floatx4* out4 = (floatx4*)outRow;
  const floatx4 z4 = {0.f, 0.f, 0.f, 0.f};

  float delta = 4.0f;                 // tail window width (raw-logit units)
  int   cnt   = 0;
  for (int attempt = 0; attempt < 16; ++attempt) {
    if (tid == 0) sCount = 0;
    __syncthreads();
    const float thrX = rawMax - delta;
    const bool first = (attempt == 0);
    for (int i = tid; i < VOCAB / 4; i += NB) {
      float4 x4 = row4[i];
      if (first) __builtin_nontemporal_store(z4, &out4[i]);  // NT: keep L2 for logits
      float xs[4] = {x4.x, x4.y, x4.z, x4.w};
#pragma unroll
      for (int c = 0; c < 4; ++c) {
        if (xs[c] >= thrX) {
          int pos = atomicAdd(&sCount, 1);
          if (pos < CAP) { candV[pos] = xs[c]; candI[pos] = i * 4 + c; }
        }
      }
    }
    __syncthreads();
    cnt = sCount;
    if (cnt > CAP)      { delta *= 0.5f; cnt = CAP; continue; }  // drops possible
    if (cnt >= kk)      break;                                   // window covers kth
    delta *= 2.0f;                                               // widen (rare)
  }

  // ---------------- bitonic sort (descending by value) ------------------
  int sortN = 64; while (sortN < cnt) sortN <<= 1;   // pow2, <= CAP
  for (int i = cnt + tid; i < sortN; i += NB) { candV[i] = -INFINITY; candI[i] = INT_MAX; }
  for (int size = 2; size <= sortN; size <<= 1) {
    for (int stride = size >> 1; stride > 0; stride >>= 1) {
      __syncthreads();
      for (int j = tid; j < sortN; j += NB) {
        int p = j ^ stride;
        if (p > j) {
          float a = candV[j], b = candV[p];
          bool dir = ((j & size) == 0);
          if ((a < b) == dir) {
            candV[j] = b; candV[p] = a;
            int ti = candI[j]; candI[j] = candI[p]; candI[p] = ti;
          }
        }
      }
    }
  }
  __syncthreads();

  // ---------------- tiny pipeline on <=64 sorted candidates -------------
  if (tid == 0) {
    int kc = kk; if (kc > cnt) kc = cnt;
    float kthVal = candV[kc - 1];
    int nk = 0;
    while (nk < 64 && nk < cnt && candV[nk] >= kthVal) nk++;  // ties kept

    float rowMax = sRowMax, rowZ = sRowZ;
    float sum1 = 0.f;
    for (int j = 0; j < nk; ++j) {
      float p = __expf(candV[j] * invT - rowMax) / rowZ;  // softmax prob
      sQ[j] = p; sIdx2[j] = candI[j]; sum1 += p;
    }
    float inv1 = 1.f / sum1;                 // top-k renorm
    float topp = topps[r];                   // top-p: prefix + value ties
    float cs = 0.f, qth = 0.f;
    for (int j = 0; j < nk; ++j) {
      float q = sQ[j] * inv1; sQ[j] = q;
      if (cs < topp) qth = q;
      cs += q;
    }
    float sum2 = 0.f;
    for (int j = 0; j < nk; ++j) if (sQ[j] >= qth) sum2 += sQ[j];
    float inv2 = 1.f / sum2;
    float rmax = sQ[0] * inv2;               // min-p
    float mp = minps[r] * rmax;
    float sum3 = 0.f;
    for (int j = 0; j < nk; ++j) {
      float rj = (sQ[j] >= qth) ? sQ[j] * inv2 : 0.f;
      sQ[j] = rj;
      if (rj > 0.f && rj >= mp) sum3 += rj;
    }
    float inv3 = 1.f / sum3;
    float tot = 0.f;
    for (int j = 0; j < nk; ++j) {
      float f = (sQ[j] > 0.f && sQ[j] >= mp) ? sQ[j] * inv3 : 0.f;
      sF[j] = f; tot += f;
    }
    sNK = nk;
    float tgt = noise[r] * tot;              // u * cdf[-1]
    sTarget = tgt;
    if (tgt <= 0.f) sTok = 0;                // reference argmax picks index 0
  }
  __syncthreads();

  // inverse-CDF sample (vocab-index order) + scatter final probs
  const int nk = sNK;
  if (tid < nk) {
    float f = sF[tid];
    if (f > 0.f) {
      int myIdx = sIdx2[tid];
      float excl = 0.f;
      for (int i = 0; i < nk; ++i)
        if (sF[i] > 0.f && sIdx2[i] < myIdx) excl += sF[i];
      if (excl + f >= sTarget) atomicMin(&sTok, myIdx);
      outRow[myIdx] = f;
    }
  }
  __syncthreads();
  if (tid == 0) {
    int t = sTok;
    if (t == INT_MAX) {            // fp-rounding fallback: last kept token
      t = 0;
      for (int j = 0; j < nk; ++j)
        if (sF[j] > 0.f && sIdx2[j] > t) t = sIdx2[j];
    }
    tokensOut[r] = t;
  }
}

extern "C" void kernel_launch(void* const* d_in, const int* in_sizes, int n_in,
                              void* d_out, int out_size, void* d_ws, size_t ws_size,
                              hipStream_t stream) {
  (void)in_sizes; (void)n_in; (void)out_size; (void)d_ws; (void)ws_size;
  const float* logits = (const float*)d_in[0];
  const float* temps  = (const float*)d_in[1];
  const int*   topks  = (const int*)d_in[2];
  const float* topps  = (const float*)d_in[3];
  const float* minps  = (const float*)d_in[4];
  const float* noise  = (const float*)d_in[5];
  int*   tokens = (int*)d_out;                  // first 256 elements (int32 bits)
  float* probs  = (float*)d_out + 256;          // then 256*128000 f32 probs
  sampler_kernel<<<dim3(256), dim3(NB), 0, stream>>>(
      logits, temps, topks, topps, minps, noise, tokens, probs);
}